// NeuralMemory_2001454760825
// MI455X (gfx1250) — compile-verified
//
#include <hip/hip_runtime.h>
#include <hip/hip_bf16.h>

// ---------------------------------------------------------------------------
// NeuralMemory for gfx1250 (MI455X): f32 WMMA (16x16x4) implementation.
// Round 4: double-buffered async global->LDS staging (prefetch next tile,
// wait asynccnt<=2) to hide HBM/L2 latency behind the per-tile WMMA chain.
// ---------------------------------------------------------------------------

typedef float v2f __attribute__((ext_vector_type(2)));
typedef float v8f __attribute__((ext_vector_type(8)));

#define ALPHA_C 0.999f
#define THETA_C 0.05f
#define EPS_C   1e-12f

#define NROWS   1048576
#define NTILES  (NROWS / 16)
#define K1_BLOCKS 256
#define K3_BLOCKS 1024
#define WAVES_PB  4
#define GSLOT 1072   // 512 (W0) + 512 (W1) + 32 (b0) + 16 (b1)

// ---- CDNA5 async global->LDS staging (ASYNCcnt) ------------------------------
#if defined(__has_builtin)
#if __has_builtin(__builtin_amdgcn_global_load_async_to_lds_b128)
#define USE_ASYNC_LDS 1
#endif
#endif

#ifdef USE_ASYNC_LDS
typedef int v4i_vs __attribute__((vector_size(16)));
typedef __attribute__((address_space(1))) v4i_vs  gv4i;
typedef __attribute__((address_space(3))) v4i_vs  lv4i;
#if __has_builtin(__builtin_amdgcn_s_wait_asynccnt)
#define ASYNC_WAIT(n) __builtin_amdgcn_s_wait_asynccnt(n)
#else
#define ASYNC_WAIT(n) asm volatile("s_wait_asynccnt %0" ::"i"(n) : "memory")
#endif
// current tile's 2 async loads were issued one iteration ago; the 2 newest
// (prefetch) loads may remain outstanding -> wait asynccnt <= 2.
#define STAGE_WAIT()                 \
  do {                               \
    ASYNC_WAIT(2);                   \
    __builtin_amdgcn_wave_barrier(); \
  } while (0)
#else
#define STAGE_WAIT() __builtin_amdgcn_wave_barrier()
#endif

__device__ __forceinline__ v8f vzero8() {
  v8f z;
#pragma unroll
  for (int j = 0; j < 8; ++j) z[j] = 0.0f;
  return z;
}

__device__ __forceinline__ v8f wmma4(v2f a, v2f b, v8f c) {
  // V_WMMA_F32_16X16X4_F32 : D = A(16x4) * B(4x16) + C(16x16)
  return __builtin_amdgcn_wmma_f32_16x16x4_f32(
      /*neg_a=*/false, a, /*neg_b=*/false, b,
      /*c_mod=*/(short)0, c, /*reuse_a=*/false, /*reuse_b=*/false);
}

// hardware-approx math: v_rcp_f32 / v_sqrt_f32 / v_exp_f32 (no v_div_* chains)
__device__ __forceinline__ float frcp(float x)  { return __builtin_amdgcn_rcpf(x); }
__device__ __forceinline__ float fsqrt(float x) { return __builtin_amdgcn_sqrtf(x); }
__device__ __forceinline__ float sigf(float x)  { return frcp(1.0f + __expf(-x)); }
__device__ __forceinline__ float siluf(float x) { return x * sigf(x); }
// silu'(x) given s = sigmoid(x)
__device__ __forceinline__ float dsilu_s(float x, float s) {
  return s * (1.0f + x * (1.0f - s));
}

// sum across the 16-lane half of a wave32 (lanes 0-15 / 16-31 independently)
__device__ __forceinline__ float half16_sum(float v) {
  v += __shfl_xor(v, 1, 32);
  v += __shfl_xor(v, 2, 32);
  v += __shfl_xor(v, 4, 32);
  v += __shfl_xor(v, 8, 32);
  return v;
}

// A-frag: A[m][k0+kk] from row-major tile T (leading dim ld); m = lane&15
// layout: VGPR0 = K {k0+2*hi}, VGPR1 = K {k0+1+2*hi}
__device__ __forceinline__ v2f fragA(const float* T, int ld, int k0, int m, int hi) {
  v2f a;
  a.x = T[m * ld + k0 + 2 * hi];
  a.y = T[m * ld + k0 + 2 * hi + 1];
  return a;
}
// A-frag of a transposed tile: A[m][k] = T[k][m]
__device__ __forceinline__ v2f fragAT(const float* T, int ld, int k0, int m, int hi) {
  v2f a;
  a.x = T[(k0 + 2 * hi) * ld + m];
  a.y = T[(k0 + 2 * hi + 1) * ld + m];
  return a;
}
// B-frag: B[k][n] = M[k][n] (row-major K x N), n = lane&15
__device__ __forceinline__ v2f fragB(const float* M, int ld, int k0, int n, int hi) {
  v2f b;
  b.x = M[(k0 + 2 * hi) * ld + n];
  b.y = M[(k0 + 2 * hi + 1) * ld + n];
  return b;
}
// B-frag of W^T where W is row-major (N x K): B[k][n] = W[n][k]
__device__ __forceinline__ v2f fragBT(const float* W, int ld, int k0, int n, int hi) {
  v2f b;
  b.x = W[n * ld + k0 + 2 * hi];
  b.y = W[n * ld + k0 + 2 * hi + 1];
  return b;
}

// Issue the copy of one 16x16 f32 tile (row-major, ld=20 padded) into LDS.
// Async path: no wait here — caller pairs with STAGE_WAIT().
__device__ __forceinline__ void stage_issue(const float* __restrict__ g,
                                            float* sX, int lane) {
#ifdef USE_ASYNC_LDS
  int e0 = lane * 4;
  int e1 = 128 + lane * 4;
  const __attribute__((address_space(1))) float* g0 =
      (const __attribute__((address_space(1))) float*)(g + lane * 4);
  const __attribute__((address_space(1))) float* g1 =
      (const __attribute__((address_space(1))) float*)(g + 128 + lane * 4);
  __attribute__((address_space(3))) float* l0 =
      (__attribute__((address_space(3))) float*)&sX[(e0 >> 4) * 20 + (e0 & 15)];
  __attribute__((address_space(3))) float* l1 =
      (__attribute__((address_space(3))) float*)&sX[(e1 >> 4) * 20 + (e1 & 15)];
  __builtin_amdgcn_global_load_async_to_lds_b128((gv4i*)g0, (lv4i*)l0, 0, 0);
  __builtin_amdgcn_global_load_async_to_lds_b128((gv4i*)g1, (lv4i*)l1, 0, 0);
#else
  const float4* gX = (const float4*)g;
  float4 xa = gX[lane * 2], xb = gX[lane * 2 + 1];
  int e = lane * 8, r = e >> 4, c0 = e & 15;
  sX[r * 20 + c0 + 0] = xa.x; sX[r * 20 + c0 + 1] = xa.y;
  sX[r * 20 + c0 + 2] = xa.z; sX[r * 20 + c0 + 3] = xa.w;
  sX[r * 20 + c0 + 4] = xb.x; sX[r * 20 + c0 + 5] = xb.y;
  sX[r * 20 + c0 + 6] = xb.z; sX[r * 20 + c0 + 7] = xb.w;
#endif
}

// ---------------------------------------------------------------------------
// Kernel 1: per-block partial gradients of (W0,b0,W1,b1).
// ---------------------------------------------------------------------------
__global__ __launch_bounds__(128) void nm_grad_kernel(
    const float* __restrict__ X,  const float* __restrict__ W0,
    const float* __restrict__ b0, const float* __restrict__ W1,
    const float* __restrict__ b1, const float* __restrict__ WV,
    const float* __restrict__ WQ, float* __restrict__ partials) {
  __shared__ float sWQ[256], sWV[256], sW0[512], sb0[32], sW1[512], sb1[16];
  __shared__ float stage[WAVES_PB * 2112];  // per-wave: X0(320) X1(320) Q(320) H(576) T(576)
  __shared__ float slots[WAVES_PB * GSLOT];

  const int tid = threadIdx.x;
  for (int i = tid; i < 256; i += 128) { sWQ[i] = WQ[i]; sWV[i] = WV[i]; }
  for (int i = tid; i < 512; i += 128) { sW0[i] = W0[i]; sW1[i] = W1[i]; }
  if (tid < 32) sb0[tid] = b0[tid];
  if (tid < 16) sb1[tid] = b1[tid];
  __syncthreads();

  const int wave = tid >> 5;
  const int lane = tid & 31;
  const int m  = lane & 15;
  const int hi = lane >> 4;
  const int n  = m;

  float* sX0 = stage + wave * 2112;  // 16x16, ld 20 (double-buffered)
  float* sX1 = sX0 + 320;
  float* sQ  = sX1 + 320;            // 16x16, ld 20
  float* sH  = sQ + 320;             // 16x32, ld 36
  float* sT  = sH + 576;             // 16x32, ld 36

  // Weight B-fragments (registers, reused across all tiles)
  v2f BQ[4], BV[4], BW0T[2][4], BW1T[8], BW1f[2][4];
#pragma unroll
  for (int c = 0; c < 4; ++c) {
    BQ[c]       = fragBT(sWQ, 16, 4 * c, n, hi);        // B = WQ^T
    BV[c]       = fragBT(sWV, 16, 4 * c, n, hi);        // B = WV^T
    BW0T[0][c]  = fragBT(sW0, 16, 4 * c, n, hi);        // B = W0^T, cols 0-15
    BW0T[1][c]  = fragBT(sW0, 16, 4 * c, n + 16, hi);   // B = W0^T, cols 16-31
    BW1f[0][c]  = fragB (sW1, 32, 4 * c, n, hi);        // B = W1,  cols 0-15
    BW1f[1][c]  = fragB (sW1, 32, 4 * c, n + 16, hi);   // B = W1,  cols 16-31
  }
#pragma unroll
  for (int c = 0; c < 8; ++c) BW1T[c] = fragBT(sW1, 32, 4 * c, n, hi);  // B = W1^T

  v8f accW0a = vzero8(), accW0b = vzero8();
  v8f accW1a = vzero8(), accW1b = vzero8();
  float gb1a = 0.f, gb0a = 0.f, gb0b = 0.f;
  const float bias0a = sb0[n], bias0b = sb0[16 + n], bias1 = sb1[n];

  const int wid     = blockIdx.x * WAVES_PB + wave;
  const int wstride = K1_BLOCKS * WAVES_PB;
  const float SCALE = 2.0f / 16777216.0f;  // 2/(N*16)

  // prime the pipeline: stage first tile into buffer 0
  stage_issue(X + (size_t)wid * 256, sX0, lane);
  int p = 0;

  for (int t = wid; t < NTILES; t += wstride) {
    // prefetch next tile into the alternate buffer (redundant-safe on last iter)
    int tn   = t + wstride;
    int tpre = (tn < NTILES) ? tn : t;
    stage_issue(X + (size_t)tpre * 256, p ? sX0 : sX1, lane);
    STAGE_WAIT();  // current buffer's loads (issued last iter) are complete
    const float* sX = p ? sX1 : sX0;
    p ^= 1;

    // ---- XQ = X @ WQ^T ; XV = X @ WV^T ----
    v8f xq = vzero8(), xv = vzero8();
#pragma unroll
    for (int c = 0; c < 4; ++c) {
      v2f aX = fragA(sX, 20, 4 * c, m, hi);
      xq = wmma4(aX, BQ[c], xq);
      xv = wmma4(aX, BV[c], xv);
    }

    // v = silu(XV); q = l2norm(silu(XQ))
    v8f vv, qs;
#pragma unroll
    for (int j = 0; j < 8; ++j) { vv[j] = siluf(xv[j]); qs[j] = siluf(xq[j]); }
#pragma unroll
    for (int j = 0; j < 8; ++j) {
      float ss  = half16_sum(qs[j] * qs[j]);
      float inv = frcp(fmaxf(fsqrt(ss), EPS_C));
      qs[j] *= inv;
    }
#pragma unroll
    for (int j = 0; j < 8; ++j) sQ[(j + 8 * hi) * 20 + n] = qs[j];
    __builtin_amdgcn_wave_barrier();

    // ---- z0 = q @ W0^T + b0 ; h = silu(z0) (cache sigmoid for backward) ----
    v8f z0a, z0b;
#pragma unroll
    for (int j = 0; j < 8; ++j) { z0a[j] = bias0a; z0b[j] = bias0b; }
#pragma unroll
    for (int c = 0; c < 4; ++c) {
      v2f aQ = fragA(sQ, 20, 4 * c, m, hi);
      z0a = wmma4(aQ, BW0T[0][c], z0a);
      z0b = wmma4(aQ, BW0T[1][c], z0b);
    }
    v8f s0a, s0b;
#pragma unroll
    for (int j = 0; j < 8; ++j) {
      s0a[j] = sigf(z0a[j]);
      s0b[j] = sigf(z0b[j]);
      sH[(j + 8 * hi) * 36 + n]      = z0a[j] * s0a[j];
      sH[(j + 8 * hi) * 36 + 16 + n] = z0b[j] * s0b[j];
    }
    __builtin_amdgcn_wave_barrier();

    // ---- z1 = h @ W1^T + b1 ; y = silu(z1) ; dz1 ----
    v8f z1;
#pragma unroll
    for (int j = 0; j < 8; ++j) z1[j] = bias1;
#pragma unroll
    for (int c = 0; c < 8; ++c) {
      v2f aH = fragA(sH, 36, 4 * c, m, hi);
      z1 = wmma4(aH, BW1T[c], z1);
    }
    v8f dz1;
#pragma unroll
    for (int j = 0; j < 8; ++j) {
      float s = sigf(z1[j]);
      float y = z1[j] * s;
      float d = SCALE * (y - vv[j]);
      dz1[j]  = d * dsilu_s(z1[j], s);
      gb1a += dz1[j];
    }
#pragma unroll
    for (int j = 0; j < 8; ++j) sT[(j + 8 * hi) * 36 + n] = dz1[j];
    __builtin_amdgcn_wave_barrier();

    // ---- gW1 += dz1^T @ h (accumulate in WMMA C regs) ----
#pragma unroll
    for (int c = 0; c < 4; ++c) {
      v2f aT = fragAT(sT, 36, 4 * c, m, hi);
      accW1a = wmma4(aT, fragB(sH, 36, 4 * c, n, hi),      accW1a);
      accW1b = wmma4(aT, fragB(sH, 36, 4 * c, n + 16, hi), accW1b);
    }

    // ---- dh = dz1 @ W1 ; dz0 = dh * silu'(z0) ----
    v8f dha = vzero8(), dhb = vzero8();
#pragma unroll
    for (int c = 0; c < 4; ++c) {
      v2f aD = fragA(sT, 36, 4 * c, m, hi);
      dha = wmma4(aD, BW1f[0][c], dha);
      dhb = wmma4(aD, BW1f[1][c], dhb);
    }
    v8f d0a, d0b;
#pragma unroll
    for (int j = 0; j < 8; ++j) {
      d0a[j] = dha[j] * dsilu_s(z0a[j], s0a[j]);
      d0b[j] = dhb[j] * dsilu_s(z0b[j], s0b[j]);
      gb0a += d0a[j];
      gb0b += d0b[j];
    }
    __builtin_amdgcn_wave_barrier();
#pragma unroll
    for (int j = 0; j < 8; ++j) {
      sT[(j + 8 * hi) * 36 + n]      = d0a[j];
      sT[(j + 8 * hi) * 36 + 16 + n] = d0b[j];
    }
    __builtin_amdgcn_wave_barrier();

    // ---- gW0 += dz0^T @ q ----
#pragma unroll
    for (int c = 0; c < 4; ++c) {
      v2f bQ2 = fragB(sQ, 20, 4 * c, n, hi);
      v2f aT0 = fragAT(sT, 36, 4 * c, m, hi);        // dz0 cols 0-15
      v2f aT1 = fragAT(sT + 16, 36, 4 * c, m, hi);   // dz0 cols 16-31
      accW0a = wmma4(aT0, bQ2, accW0a);
      accW0b = wmma4(aT1, bQ2, accW0b);
    }
  }

  // ---- per-wave grads -> LDS slot ----
  float* slot = slots + wave * GSLOT;
#pragma unroll
  for (int j = 0; j < 8; ++j) {
    int mm = j + 8 * hi;
    slot[mm * 16 + n]              = accW0a[j];   // gW0 rows 0-15
    slot[(16 + mm) * 16 + n]       = accW0b[j];   // gW0 rows 16-31
    slot[512 + mm * 32 + n]        = accW1a[j];   // gW1 cols 0-15
    slot[512 + mm * 32 + 16 + n]   = accW1b[j];   // gW1 cols 16-31
  }
  float t0 = gb0a + __shfl_xor(gb0a, 16, 32);
  float t1 = gb0b + __shfl_xor(gb0b, 16, 32);
  float t2 = gb1a + __shfl_xor(gb1a, 16, 32);
  if (hi == 0) {
    slot[1024 + n] = t0;
    slot[1040 + n] = t1;
    slot[1056 + n] = t2;
  }
  __syncthreads();

  // fixed-order cross-wave sum -> per-block partial (deterministic)
  for (int i = tid; i < GSLOT; i += 128) {
    float s = slots[i] + slots[GSLOT + i] + slots[2 * GSLOT + i] + slots[3 * GSLOT + i];
    partials[(size_t)blockIdx.x * GSLOT + i] = s;
  }
}

// ---------------------------------------------------------------------------
// Kernel 2: reduce partials (fixed order) and form new params.
// One wave per parameter index. grid = GSLOT/8 blocks x 256 threads.
// ---------------------------------------------------------------------------
__global__ __launch_bounds__(256) void nm_reduce_kernel(
    const float* __restrict__ partials,
    const float* __restrict__ W0, const float* __restrict__ b0,
    const float* __restrict__ W1, const float* __restrict__ b1,
    float* __restrict__ params) {
  const int wave = threadIdx.x >> 5;
  const int lane = threadIdx.x & 31;
  const int i = blockIdx.x * 8 + wave;
  if (i >= GSLOT) return;
  float g = 0.0f;
  for (int k = lane; k < K1_BLOCKS; k += 32) g += partials[(size_t)k * GSLOT + i];
  g += __shfl_xor(g, 1, 32);
  g += __shfl_xor(g, 2, 32);
  g += __shfl_xor(g, 4, 32);
  g += __shfl_xor(g, 8, 32);
  g += __shfl_xor(g, 16, 32);
  if (lane == 0) {
    float p;
    if (i < 512)       p = W0[i];
    else if (i < 1024) p = W1[i - 512];
    else if (i < 1056) p = b0[i - 1024];
    else               p = b1[i - 1056];
    params[i] = ALPHA_C * p - THETA_C * g;
  }
}

// ---------------------------------------------------------------------------
// Kernel 3: retrieve — y = mlp(new_params, silu(l2norm(X @ WQ^T)))
// ---------------------------------------------------------------------------
__global__ __launch_bounds__(128) void nm_retrieve_kernel(
    const float* __restrict__ X, const float* __restrict__ WQ,
    const float* __restrict__ params, float* __restrict__ out) {
  __shared__ float sWQ[256], sW0[512], sb0[32], sW1[512], sb1[16];
  __shared__ float stage[WAVES_PB * 1536];  // per-wave: X0(320) X1(320) R(320) H(576)

  const int tid = threadIdx.x;
  for (int i = tid; i < 256; i += 128) sWQ[i] = WQ[i];
  for (int i = tid; i < 512; i += 128) { sW0[i] = params[i]; sW1[i] = params[512 + i]; }
  if (tid < 32) sb0[tid] = params[1024 + tid];
  if (tid < 16) sb1[tid] = params[1056 + tid];
  __syncthreads();

  const int wave = tid >> 5;
  const int lane = tid & 31;
  const int m  = lane & 15;
  const int hi = lane >> 4;
  const int n  = m;

  float* sX0 = stage + wave * 1536;
  float* sX1 = sX0 + 320;
  float* sR  = sX1 + 320;
  float* sH  = sR + 320;

  v2f BQ[4], BW0T[2][4], BW1T[8];
#pragma unroll
  for (int c = 0; c < 4; ++c) {
    BQ[c]      = fragBT(sWQ, 16, 4 * c, n, hi);
    BW0T[0][c] = fragBT(sW0, 16, 4 * c, n, hi);
    BW0T[1][c] = fragBT(sW0, 16, 4 * c, n + 16, hi);
  }
#pragma unroll
  for (int c = 0; c < 8; ++c) BW1T[c] = fragBT(sW1, 32, 4 * c, n, hi);

  const float bias0a = sb0[n], bias0b = sb0[16 + n], bias1 = sb1[n];

  const int wid     = blockIdx.x * WAVES_PB + wave;
  const int wstride = K3_BLOCKS * WAVES_PB;

  stage_issue(X + (size_t)wid * 256, sX0, lane);
  int p = 0;

  for (int t = wid; t < NTILES; t += wstride) {
    int tn   = t + wstride;
    int tpre = (tn < NTILES) ? tn : t;
    stage_issue(X + (size_t)tpre * 256, p ? sX0 : sX1, lane);
    STAGE_WAIT();
    const float* sX = p ? sX1 : sX0;
    p ^= 1;

    v8f xq = vzero8();
#pragma unroll
    for (int c = 0; c < 4; ++c)
      xq = wmma4(fragA(sX, 20, 4 * c, m, hi), BQ[c], xq);

    // r = silu(l2norm(xq))  (normalize FIRST, then silu)
#pragma unroll
    for (int j = 0; j < 8; ++j) {
      float ss  = half16_sum(xq[j] * xq[j]);
      float inv = frcp(fmaxf(fsqrt(ss), EPS_C));
      float tq  = xq[j] * inv;
      sR[(j + 8 * hi) * 20 + n] = siluf(tq);
    }
    __builtin_amdgcn_wave_barrier();

    v8f z0a, z0b;
#pragma unroll
    for (int j = 0; j < 8; ++j) { z0a[j] = bias0a; z0b[j] = bias0b; }
#pragma unroll
    for (int c = 0; c < 4; ++c) {
      v2f aR = fragA(sR, 20, 4 * c, m, hi);
      z0a = wmma4(aR, BW0T[0][c], z0a);
      z0b = wmma4(aR, BW0T[1][c], z0b);
    }
#pragma unroll
    for (int j = 0; j < 8; ++j) {
      sH[(j + 8 * hi) * 36 + n]      = siluf(z0a[j]);
      sH[(j + 8 * hi) * 36 + 16 + n] = siluf(z0b[j]);
    }
    __builtin_amdgcn_wave_barrier();

    v8f z1;
#pragma unroll
    for (int j = 0; j < 8; ++j) z1[j] = bias1;
#pragma unroll
    for (int c = 0; c < 8; ++c) {
      v2f aH = fragA(sH, 36, 4 * c, m, hi);
      z1 = wmma4(aH, BW1T[c], z1);
    }

    float* o = out + (size_t)t * 256;
#pragma unroll
    for (int j = 0; j < 8; ++j) o[(j + 8 * hi) * 16 + n] = siluf(z1[j]);
  }
}

// ---------------------------------------------------------------------------
extern "C" void kernel_launch(void* const* d_in, const int* in_sizes, int n_in,
                              void* d_out, int out_size, void* d_ws, size_t ws_size,
                              hipStream_t stream) {
  (void)in_sizes; (void)n_in; (void)out_size; (void)ws_size;
  // setup_inputs order: X, W0, b0, W1, b1, WK(unused), WV, WQ
  const float* X  = (const float*)d_in[0];
  const float* W0 = (const float*)d_in[1];
  const float* b0 = (const float*)d_in[2];
  const float* W1 = (const float*)d_in[3];
  const float* b1 = (const float*)d_in[4];
  const float* WV = (const float*)d_in[6];
  const float* WQ = (const float*)d_in[7];
  float* out = (float*)d_out;

  float* partials = (float*)d_ws;                              // K1_BLOCKS * GSLOT floats
  float* params   = partials + (size_t)K1_BLOCKS * GSLOT;      // GSLOT floats

  nm_grad_kernel<<<K1_BLOCKS, 128, 0, stream>>>(X, W0, b0, W1, b1, WV, WQ, partials);
  nm_reduce_kernel<<<GSLOT / 8, 256, 0, stream>>>(partials, W0, b0, W1, b1, params);
  nm_retrieve_kernel<<<K3_BLOCKS, 128, 0, stream>>>(X, WQ, params, out);
}